// BayesianSkipgram_18614388261031
// MI455X (gfx1250) — compile-verified
//
#include <hip/hip_runtime.h>
#include <hip/hip_bf16.h>
#include <math.h>

// ---------------------------------------------------------------------------
// BayesianSkipgram forward on MI455X (gfx1250, wave32, WMMA)
//   VOCAB=50257, E=1024, CTX=1024, C=50, NEG=10
// ---------------------------------------------------------------------------

#define E_DIM   1024
#define CTX_DIM 1024
#define C_LEN   50
#define NEG_N   10
#define NROWS   (1 + C_LEN + C_LEN * NEG_N)   // 551 KL rows

typedef float v2f __attribute__((ext_vector_type(2)));
typedef float v8f __attribute__((ext_vector_type(8)));

// ---------------------------------------------------------------------------
// Kernel 1: R[0..63, 0..1023] = emb[gather] @ M_w^T + M_b  via f32 WMMA.
//   row 0 -> x, rows 1..50 -> context, rows 51..63 -> padding (row of x).
// One wave (32 threads) per 16-wide column tile; 4 row-tiles per wave so the
// B fragment (M_w) is loaded once and reused 4x.
//
// V_WMMA_F32_16X16X4_F32 fragment layouts (ISA 7.12.2):
//   A (16x4, MxK): lane L<16 holds A[M=L, K=0..1] in v0..v1; lane 16+L holds K=2..3.
//   B (4x16, KxN): lane L<16 holds B[K=0..1, N=L]; lane 16+L holds K=2..3, N=L.
//   D (16x16):     VGPR j, lane L<16 -> D[M=j, N=L]; lane 16+L -> D[M=8+j, N=L].
// ---------------------------------------------------------------------------
__global__ __launch_bounds__(32) void k_gemm_R(
    const float* __restrict__ emb, const float* __restrict__ Mw,
    const float* __restrict__ Mb, const int* __restrict__ x,
    const int* __restrict__ ctx, float* __restrict__ R)
{
    const int lane  = threadIdx.x;            // 0..31, full wave (EXEC all ones)
    const int laneM = lane & 15;
    const int kHalf = (lane >> 4) * 2;        // 0 for lanes 0-15, 2 for 16-31
    const int n0    = blockIdx.x * 16;        // output column tile base

    // Gathered vocab ids for the 4 row tiles handled by this wave.
    const float* aptr[4];
#pragma unroll
    for (int t = 0; t < 4; ++t) {
        int r   = t * 16 + laneM;
        int vid = (r == 0) ? x[0] : ((r <= C_LEN) ? ctx[r - 1] : x[0]);
        aptr[t] = emb + (size_t)vid * E_DIM + kHalf;
    }
    const float* bptr = Mw + (size_t)(n0 + laneM) * E_DIM + kHalf;

    v8f acc[4];
#pragma unroll
    for (int t = 0; t < 4; ++t) acc[t] = (v8f){};

    for (int k0 = 0; k0 < E_DIM; k0 += 4) {
        // Stream M_w: hint next chunk into cache (global_prefetch_b8).
        __builtin_prefetch(bptr + k0 + 128, 0, 0);
        v2f b  = *(const v2f*)(bptr + k0);
        v2f a0 = *(const v2f*)(aptr[0] + k0);
        v2f a1 = *(const v2f*)(aptr[1] + k0);
        v2f a2 = *(const v2f*)(aptr[2] + k0);
        v2f a3 = *(const v2f*)(aptr[3] + k0);
        acc[0] = __builtin_amdgcn_wmma_f32_16x16x4_f32(false, a0, false, b,
                                                       (short)0, acc[0], false, false);
        acc[1] = __builtin_amdgcn_wmma_f32_16x16x4_f32(false, a1, false, b,
                                                       (short)0, acc[1], false, false);
        acc[2] = __builtin_amdgcn_wmma_f32_16x16x4_f32(false, a2, false, b,
                                                       (short)0, acc[2], false, false);
        acc[3] = __builtin_amdgcn_wmma_f32_16x16x4_f32(false, a3, false, b,
                                                       (short)0, acc[3], false, false);
    }

    const float bias   = Mb[n0 + laneM];
    const int   rowOff = (lane >> 4) * 8;     // lanes 16-31 hold M=8..15 of tile
#pragma unroll
    for (int t = 0; t < 4; ++t) {
#pragma unroll
        for (int j = 0; j < 8; ++j) {
            int grow = t * 16 + rowOff + j;
            R[(size_t)grow * CTX_DIM + n0 + laneM] = acc[t][j] + bias;
        }
    }
}

// ---------------------------------------------------------------------------
// Kernel 2: h[0..1023] = C * relu(Rw), h[1024..2047] = sum_c relu(Rc[c])
// ---------------------------------------------------------------------------
__global__ __launch_bounds__(256) void k_h(const float* __restrict__ R,
                                           float* __restrict__ h)
{
    int tid = blockIdx.x * blockDim.x + threadIdx.x;   // 0..2047
    if (tid < CTX_DIM) {
        h[tid] = (float)C_LEN * fmaxf(R[tid], 0.0f);
    } else {
        int j = tid - CTX_DIM;
        float s = 0.0f;
        for (int c = 1; c <= C_LEN; ++c)
            s += fmaxf(R[(size_t)c * CTX_DIM + j], 0.0f);
        h[tid] = s;
    }
}

// ---------------------------------------------------------------------------
// Kernel 3: mu = U_w@h + U_b ; sigma = softplus(W_w@h + W_b) ; logsig = log(sigma)
// One wave per output row n; both dot products computed in the same pass.
// ---------------------------------------------------------------------------
__global__ __launch_bounds__(128) void k_mu_sigma(
    const float* __restrict__ Uw, const float* __restrict__ Ub,
    const float* __restrict__ Ww, const float* __restrict__ Wb,
    const float* __restrict__ h, float* __restrict__ mu,
    float* __restrict__ sigma, float* __restrict__ logsig)
{
    int wave = threadIdx.x >> 5;
    int lane = threadIdx.x & 31;
    int n    = blockIdx.x * 4 + wave;          // grid covers exactly 1024 rows
    const float* ur = Uw + (size_t)n * (2 * CTX_DIM);
    const float* wr = Ww + (size_t)n * (2 * CTX_DIM);
    float su = 0.0f, sw = 0.0f;
    for (int j = lane; j < 2 * CTX_DIM; j += 32) {
        float hv = h[j];
        su = fmaf(ur[j], hv, su);
        sw = fmaf(wr[j], hv, sw);
    }
#pragma unroll
    for (int m = 16; m; m >>= 1) {
        su += __shfl_xor(su, m, 32);
        sw += __shfl_xor(sw, m, 32);
    }
    if (lane == 0) {
        float sr = sw + Wb[n];
        // numerically stable softplus
        float sg = fmaxf(sr, 0.0f) + log1pf(expf(-fabsf(sr)));
        mu[n]     = su + Ub[n];
        sigma[n]  = sg;
        logsig[n] = logf(sg);
    }
}

// ---------------------------------------------------------------------------
// Kernel 4: raw KL sums S[r] = sum_j ((sigma_j + (mu_j - m_j)^2)/s2_j + log s2_j)
//   r = 0        -> x           (kl_prior)
//   r = 1..50    -> context     (kl_pos)
//   r = 51..550  -> neg_samples (kl_neg, flattened [C,NEG])
// One wave per row.
// ---------------------------------------------------------------------------
__global__ __launch_bounds__(128) void k_kl(
    const float* __restrict__ pm, const float* __restrict__ ps,
    const float* __restrict__ mu, const float* __restrict__ sigma,
    const int* __restrict__ x, const int* __restrict__ ctx,
    const int* __restrict__ neg, float* __restrict__ S)
{
    int wave = threadIdx.x >> 5;
    int lane = threadIdx.x & 31;
    int r    = blockIdx.x * 4 + wave;
    if (r >= NROWS) return;
    int v = (r == 0) ? x[0] : ((r <= C_LEN) ? ctx[r - 1] : neg[r - (C_LEN + 1)]);
    const float* pmr = pm + (size_t)v * CTX_DIM;
    const float* psr = ps + (size_t)v * CTX_DIM;
    float s = 0.0f;
    for (int j = lane; j < CTX_DIM; j += 32) {
        float d  = mu[j] - pmr[j];
        float sv = psr[j];
        float s2 = sv * sv;
        s += (sigma[j] + d * d) / s2 + logf(s2);
    }
#pragma unroll
    for (int m = 16; m; m >>= 1) s += __shfl_xor(s, m, 32);
    if (lane == 0) S[r] = s;
}

// ---------------------------------------------------------------------------
// Kernel 5: single-block deterministic finish.
//   logdet = sum log(sigma)
//   kl(r)  = 0.5*(S[r] - k - logdet);  kl_neg - kl_pos = 0.5*(S_neg - S_pos)
//   out    = sum max(0, 0.5*(S_neg - S_pos) + 1) - 0.5*(S[0] - k - logdet)
// ---------------------------------------------------------------------------
__global__ __launch_bounds__(256) void k_final(
    const float* __restrict__ logsig, const float* __restrict__ S,
    float* __restrict__ out)
{
    __shared__ float red[256];
    int t = threadIdx.x;

    float ld = 0.0f;
    for (int j = t; j < CTX_DIM; j += 256) ld += logsig[j];
    red[t] = ld;
    __syncthreads();
#pragma unroll
    for (int s = 128; s; s >>= 1) {
        if (t < s) red[t] += red[t + s];
        __syncthreads();
    }
    float logdet = red[0];
    __syncthreads();

    float lik = 0.0f;
    for (int i = t; i < C_LEN * NEG_N; i += 256) {
        int   c = i / NEG_N;
        float d = 0.5f * (S[1 + C_LEN + i] - S[1 + c]) + 1.0f;
        lik += fmaxf(d, 0.0f);
    }
    red[t] = lik;
    __syncthreads();
#pragma unroll
    for (int s = 128; s; s >>= 1) {
        if (t < s) red[t] += red[t + s];
        __syncthreads();
    }
    if (t == 0) {
        float kl_prior = 0.5f * (S[0] - (float)CTX_DIM - logdet);
        out[0] = red[0] - kl_prior;
    }
}

// ---------------------------------------------------------------------------
// launch
// ---------------------------------------------------------------------------
extern "C" void kernel_launch(void* const* d_in, const int* in_sizes, int n_in,
                              void* d_out, int out_size, void* d_ws, size_t ws_size,
                              hipStream_t stream)
{
    const int*   x    = (const int*)d_in[0];
    const int*   ctx  = (const int*)d_in[1];
    const int*   neg  = (const int*)d_in[2];
    const float* emb  = (const float*)d_in[3];
    const float* Mw   = (const float*)d_in[4];
    const float* Mb   = (const float*)d_in[5];
    const float* Uw   = (const float*)d_in[6];
    const float* Ub   = (const float*)d_in[7];
    const float* Ww   = (const float*)d_in[8];
    const float* Wb   = (const float*)d_in[9];
    const float* pmus = (const float*)d_in[10];
    const float* psig = (const float*)d_in[11];

    float* ws     = (float*)d_ws;
    float* R      = ws;                         // 64 * 1024
    float* h      = R + 64 * CTX_DIM;           // 2048
    float* mu     = h + 2 * CTX_DIM;            // 1024
    float* sigma  = mu + CTX_DIM;               // 1024
    float* logsig = sigma + CTX_DIM;            // 1024
    float* S      = logsig + CTX_DIM;           // 551

    // 1) R = gather(emb) @ M_w^T + M_b   (WMMA f32)
    k_gemm_R<<<dim3(CTX_DIM / 16), dim3(32), 0, stream>>>(emb, Mw, Mb, x, ctx, R);
    // 2) h
    k_h<<<dim3(8), dim3(256), 0, stream>>>(R, h);
    // 3) mu, sigma, log(sigma)
    k_mu_sigma<<<dim3(CTX_DIM / 4), dim3(128), 0, stream>>>(Uw, Ub, Ww, Wb, h,
                                                            mu, sigma, logsig);
    // 4) KL row sums (551 rows, 4 waves/block)
    k_kl<<<dim3((NROWS + 3) / 4), dim3(128), 0, stream>>>(pmus, psig, mu, sigma,
                                                          x, ctx, neg, S);
    // 5) final scalar
    k_final<<<dim3(1), dim3(256), 0, stream>>>(logsig, S, (float*)d_out);
}